// MyPool2d_72189810311235
// MI455X (gfx1250) — compile-verified
//
#include <hip/hip_runtime.h>
#include <stdint.h>

// 2x2 non-overlapping max pool, float32.
// x: (16, 64, 512, 512) -> out: (16, 64, 256, 256)
// Memory-bound: ~1.25 GiB traffic, floor ~57us @ 23.3 TB/s.
// Strategy: async global->LDS double-buffered pipeline (CDNA5 ASYNCcnt path),
// 128-bit coalesced loads, LDS b128 reads, non-temporal coalesced stores.

#define BLOCK 256
#define IN_W 512
#define OUT_W 256
#define TILE_OROWS 8                         // output rows per tile
#define IN_TILE_FLOATS (TILE_OROWS * 2 * IN_W)   // 8192 floats = 32 KiB
#define LOADS_PER_THREAD 8                   // 32768 B / 16 B / 256 thr
#define TILES_PER_PLANE (OUT_W / TILE_OROWS) // 32
#define N_PLANES 1024                        // 16*64
#define N_TILES (N_PLANES * TILES_PER_PLANE) // 32768
#define GRID_BLOCKS 4096                     // 8 tiles per persistent block

typedef float v2f __attribute__((ext_vector_type(2)));

__global__ __launch_bounds__(BLOCK)
void maxpool2x2_async_kernel(const float* __restrict__ in,
                             float* __restrict__ out,
                             int n_tiles) {
  __shared__ __align__(16) float lds[2][IN_TILE_FLOATS];   // 64 KiB double buffer

  const int tid = threadIdx.x;
  const uint64_t sbase = (uint64_t)(uintptr_t)in;          // SGPR pair base for GVS mode
  const unsigned ldsb0 = (unsigned)(uintptr_t)(&lds[0][0]);
  const unsigned ldsb1 = (unsigned)(uintptr_t)(&lds[1][0]);

  // Issue 8 async 128-bit global->LDS loads for one 32 KiB input tile.
  auto prefetch = [&](int tile, unsigned ldsBase) {
    const unsigned plane = (unsigned)tile / TILES_PER_PLANE;
    const unsigned frag  = (unsigned)tile % TILES_PER_PLANE;
    // byte offset of tile start inside the input tensor (max ~1.07e9, fits u32)
    const unsigned gbase = plane * (unsigned)(IN_W * IN_W * 4)
                         + frag  * (unsigned)(IN_TILE_FLOATS * 4);
#pragma unroll
    for (int t = 0; t < LOADS_PER_THREAD; ++t) {
      const unsigned j     = (unsigned)(t * BLOCK + tid);  // 0..2047, lane-consecutive
      const unsigned gofs  = gbase + j * 16u;
      const unsigned laddr = ldsBase + j * 16u;
      asm volatile("global_load_async_to_lds_b128 %0, %1, %2"
                   :: "v"(laddr), "v"(gofs), "s"(sbase)
                   : "memory");
    }
  };

  // Consume one staged tile: each thread produces 8 outputs (4 x float2).
  auto compute = [&](int tile, const float* __restrict__ buf) {
    const unsigned plane = (unsigned)tile / TILES_PER_PLANE;
    const unsigned frag  = (unsigned)tile % TILES_PER_PLANE;
    float* __restrict__ outp = out + (size_t)plane * (OUT_W * OUT_W)
                                   + (size_t)frag * (TILE_OROWS * OUT_W);
#pragma unroll
    for (int t = 0; t < 4; ++t) {
      const unsigned p    = (unsigned)(t * BLOCK + tid);   // 0..1023 output pairs
      const unsigned orow = p >> 7;                        // 0..7
      const unsigned oc2  = p & 127u;                      // pair-of-columns index
      const float4 a = *(const float4*)&buf[(2 * orow)     * IN_W + 4 * oc2];
      const float4 b = *(const float4*)&buf[(2 * orow + 1) * IN_W + 4 * oc2];
      v2f r;
      r.x = fmaxf(fmaxf(a.x, a.y), fmaxf(b.x, b.y));
      r.y = fmaxf(fmaxf(a.z, a.w), fmaxf(b.z, b.w));
      // write-once output stream: non-temporal, 8B coalesced
      __builtin_nontemporal_store(r, (v2f*)&outp[orow * OUT_W + 2 * oc2]);
    }
  };

  int tile = blockIdx.x;
  const int stride = gridDim.x;
  int buf = 0;

  if (tile < n_tiles) prefetch(tile, ldsb0);

  for (; tile < n_tiles; tile += stride) {
    const int next = tile + stride;
    if (next < n_tiles) {
      prefetch(next, buf ? ldsb0 : ldsb1);
      // 8 new loads outstanding for 'next'; wait until the 8 for 'tile' landed
      asm volatile("s_wait_asynccnt 8" ::: "memory");
    } else {
      asm volatile("s_wait_asynccnt 0" ::: "memory");
    }
    __syncthreads();                       // all waves' async data visible
    compute(tile, buf ? &lds[1][0] : &lds[0][0]);
    __syncthreads();                       // done reading before next overwrite
    buf ^= 1;
  }
}

extern "C" void kernel_launch(void* const* d_in, const int* in_sizes, int n_in,
                              void* d_out, int out_size, void* d_ws, size_t ws_size,
                              hipStream_t stream) {
  const float* x = (const float*)d_in[0];
  float* out = (float*)d_out;
  (void)in_sizes; (void)n_in; (void)out_size; (void)d_ws; (void)ws_size;
  hipLaunchKernelGGL(maxpool2x2_async_kernel,
                     dim3(GRID_BLOCKS), dim3(BLOCK), 0, stream,
                     x, out, N_TILES);
}